// MatchingLayer_63651415327148
// MI455X (gfx1250) — compile-verified
//
#include <hip/hip_runtime.h>
#include <hip/hip_bf16.h>

typedef __attribute__((ext_vector_type(2))) float v2f;
typedef __attribute__((ext_vector_type(8))) float v8f;

constexpr int   kS  = 256;   // sequence length
constexpr int   kB  = 128;   // batch
constexpr int   kD2 = 512;   // full feature dim
constexpr int   kDH = 256;   // half feature dim
constexpr int   kL  = 8;     // perspectives
constexpr float kEPS = 1e-8f;

// workspace layout (float offsets)
constexpr size_t WS_ALPHA = 0;                                   // [B][S][S]      8388608
constexpr size_t WS_HMEAN = WS_ALPHA + (size_t)kB * kS * kS;     // [S][B][DH]     8388608
constexpr size_t WS_RSUM  = WS_HMEAN + (size_t)kS * kB * kDH;    // [S][B]         32768
constexpr size_t WS_IDX   = WS_RSUM  + (size_t)kS * kB;          // [S][B] int     32768
constexpr size_t WS_NPL   = WS_IDX   + (size_t)kS * kB;          // [2][S][B]      65536
constexpr size_t WS_NQL   = WS_NPL   + (size_t)2 * kS * kB;      // [2][S][B]      65536
constexpr size_t WS_PNW   = WS_NQL   + (size_t)2 * kS * kB;      // [2][L][S][B]   524288
constexpr size_t WS_QNW   = WS_PNW   + (size_t)2 * kL * kS * kB; // [2][L][S][B]   524288

__device__ __forceinline__ v8f wmma_f32(v2f a, v2f b, v8f c) {
  // D = A(16x4 f32) * B(4x16 f32) + C(16x16 f32)
  return __builtin_amdgcn_wmma_f32_16x16x4_f32(
      /*neg_a=*/false, a, /*neg_b=*/false, b,
      /*c_mod=*/(short)0, c, /*reuse_a=*/false, /*reuse_b=*/false);
}

__device__ __forceinline__ float wave_sum(float v) {
  for (int off = 1; off < 32; off <<= 1) v += __shfl_xor(v, off, 32);
  return v;
}

// ---------------------------------------------------------------------------
// Kernel 1: plain norms of p/q halves + W2/W3-weighted norms (for maxpool)
// ---------------------------------------------------------------------------
__global__ __launch_bounds__(256) void norms_kernel(
    const float* __restrict__ p, const float* __restrict__ q,
    const float* __restrict__ W, float* __restrict__ npl,
    float* __restrict__ nql, float* __restrict__ pnw, float* __restrict__ qnw) {
  const int s = blockIdx.x, b = blockIdx.y;
  const int tid = threadIdx.x, wv = tid >> 5, lane = tid & 31;
  __shared__ float pr[kD2], qr[kD2];
  for (int i = tid; i < kD2; i += 256) {
    pr[i] = p[((size_t)s * kB + b) * kD2 + i];
    qr[i] = q[((size_t)s * kB + b) * kD2 + i];
  }
  __syncthreads();

  // wave wv handles perspective l = wv (weighted norms with W[2] fw, W[3] bw)
  float spf = 0.f, spb = 0.f, sqf = 0.f, sqb = 0.f;
  for (int j = 0; j < 8; ++j) {
    const int d = lane + j * 32;
    float w2 = W[(2 * kL + wv) * kDH + d]; w2 *= w2;
    float w3 = W[(3 * kL + wv) * kDH + d]; w3 *= w3;
    const float pf = pr[d], pb = pr[kDH + d], qf = qr[d], qb = qr[kDH + d];
    spf += pf * pf * w2; spb += pb * pb * w3;
    sqf += qf * qf * w2; sqb += qb * qb * w3;
  }
  spf = wave_sum(spf); spb = wave_sum(spb);
  sqf = wave_sum(sqf); sqb = wave_sum(sqb);
  if (lane == 0) {
    pnw[((size_t)(0 * kL + wv) * kS + s) * kB + b] = sqrtf(spf);
    pnw[((size_t)(1 * kL + wv) * kS + s) * kB + b] = sqrtf(spb);
    qnw[((size_t)(0 * kL + wv) * kS + s) * kB + b] = sqrtf(sqf);
    qnw[((size_t)(1 * kL + wv) * kS + s) * kB + b] = sqrtf(sqb);
  }
  // plain norms on wave 0
  if (wv == 0) {
    float a0 = 0.f, a1 = 0.f, a2 = 0.f, a3 = 0.f;
    for (int j = 0; j < 8; ++j) {
      const int d = lane + j * 32;
      a0 += pr[d] * pr[d];           a1 += pr[kDH + d] * pr[kDH + d];
      a2 += qr[d] * qr[d];           a3 += qr[kDH + d] * qr[kDH + d];
    }
    a0 = wave_sum(a0); a1 = wave_sum(a1); a2 = wave_sum(a2); a3 = wave_sum(a3);
    if (lane == 0) {
      npl[(size_t)0 * kS * kB + s * kB + b] = sqrtf(a0);
      npl[(size_t)1 * kS * kB + s * kB + b] = sqrtf(a1);
      nql[(size_t)0 * kS * kB + s * kB + b] = sqrtf(a2);
      nql[(size_t)1 * kS * kB + s * kB + b] = sqrtf(a3);
    }
  }
}

// ---------------------------------------------------------------------------
// Kernel 2: m_full (channels 0-7 fw with q_fw[-1]/W0, 8-15 bw with q_bw[0]/W1)
// ---------------------------------------------------------------------------
__global__ __launch_bounds__(256) void full_match_kernel(
    const float* __restrict__ p, const float* __restrict__ q,
    const float* __restrict__ W, float* __restrict__ out) {
  const int s = blockIdx.x, b = blockIdx.y;
  const int tid = threadIdx.x, wv = tid >> 5, lane = tid & 31;
  __shared__ float pr[kD2], qref[kD2];
  for (int i = tid; i < kD2; i += 256)
    pr[i] = p[((size_t)s * kB + b) * kD2 + i];
  // qref[0:256] = q_fw[last], qref[256:512] = q_bw[first]
  qref[tid]       = q[((size_t)(kS - 1) * kB + b) * kD2 + tid];
  qref[kDH + tid] = q[((size_t)0 * kB + b) * kD2 + kDH + tid];
  __syncthreads();

  const int l = wv;
  float nf = 0.f, af = 0.f, bf = 0.f, nb = 0.f, ab = 0.f, bb = 0.f;
  for (int j = 0; j < 8; ++j) {
    const int d = lane + j * 32;
    float w0 = W[(0 * kL + l) * kDH + d]; w0 *= w0;
    float w1 = W[(1 * kL + l) * kDH + d]; w1 *= w1;
    const float pf = pr[d], qf = qref[d];
    nf += pf * qf * w0; af += pf * pf * w0; bf += qf * qf * w0;
    const float pb = pr[kDH + d], qb = qref[kDH + d];
    nb += pb * qb * w1; ab += pb * pb * w1; bb += qb * qb * w1;
  }
  nf = wave_sum(nf); af = wave_sum(af); bf = wave_sum(bf);
  nb = wave_sum(nb); ab = wave_sum(ab); bb = wave_sum(bb);
  if (lane == 0) {
    float* o = out + ((size_t)s * kB + b) * 64;
    o[l]      = nf / fmaxf(sqrtf(af) * sqrtf(bf), kEPS);
    o[8 + l]  = nb / fmaxf(sqrtf(ab) * sqrtf(bb), kEPS);
  }
}

// ---------------------------------------------------------------------------
// Kernel 3: maxpool match (channels 16-23 fw, 24-31 bw) — WMMA S = (p∘W²)Qᵀ
// ---------------------------------------------------------------------------
__global__ __launch_bounds__(128) void maxpool_kernel(
    const float* __restrict__ p, const float* __restrict__ q,
    const float* __restrict__ W, const float* __restrict__ pnw,
    const float* __restrict__ qnw, float* __restrict__ out) {
  const int b = blockIdx.x, l = blockIdx.y;
  const int sblk = blockIdx.z & 3, dir = blockIdx.z >> 2;
  const int dirOff = dir * kDH;
  const int tid = threadIdx.x;
  const int wv = tid >> 5, lane = tid & 31, lane16 = lane & 15, kh = lane >> 4;

  __shared__ float w2sq[kDH];
  __shared__ float pl[64 * 68];
  __shared__ float ql[128 * 68];

  for (int i = tid; i < kDH; i += 128) {
    float w = W[((2 + dir) * kL + l) * kDH + i];
    w2sq[i] = w * w;
  }

  const int rbase = sblk * 64 + wv * 16 + kh * 8;  // first C-row for this lane
  float n1v[8];
  for (int r = 0; r < 8; ++r)
    n1v[r] = pnw[((size_t)(dir * kL + l) * kS + (rbase + r)) * kB + b];

  float rmax[8];
  for (int r = 0; r < 8; ++r) rmax[r] = -3.4e38f;

  const int aRowLds = (wv * 16 + lane16) * 68;     // A-fragment LDS row base

  for (int tb = 0; tb < 2; ++tb) {
    v8f acc[8];
    const v8f vzero = {0.f, 0.f, 0.f, 0.f, 0.f, 0.f, 0.f, 0.f};
    for (int tt = 0; tt < 8; ++tt) acc[tt] = vzero;

    for (int kc = 0; kc < 4; ++kc) {
      __syncthreads();
      for (int i = tid; i < 64 * 64; i += 128) {        // p chunk, W²-scaled
        const int r = i >> 6, k = i & 63, kk = kc * 64 + k;
        pl[r * 68 + k] =
            p[((size_t)(sblk * 64 + r) * kB + b) * kD2 + dirOff + kk] * w2sq[kk];
      }
      for (int i = tid; i < 128 * 64; i += 128) {       // q chunk, raw
        const int r = i >> 6, k = i & 63;
        ql[r * 68 + k] =
            q[((size_t)(tb * 128 + r) * kB + b) * kD2 + dirOff + kc * 64 + k];
      }
      __syncthreads();
      for (int tt = 0; tt < 8; ++tt) {
        const int bRowLds = (tt * 16 + lane16) * 68;
        v8f c = acc[tt];
        for (int k0 = 0; k0 < 64; k0 += 4) {
          v2f a  = *(const v2f*)&pl[aRowLds + k0 + 2 * kh];
          v2f bv = *(const v2f*)&ql[bRowLds + k0 + 2 * kh];
          c = wmma_f32(a, bv, c);
        }
        acc[tt] = c;
      }
    }
    // finalize this t-block: normalize + running max
    for (int tt = 0; tt < 8; ++tt) {
      const int t = tb * 128 + tt * 16 + lane16;
      const float n2 = qnw[((size_t)(dir * kL + l) * kS + t) * kB + b];
      for (int r = 0; r < 8; ++r) {
        const float sim = acc[tt][r] / fmaxf(n1v[r] * n2, kEPS);
        rmax[r] = fmaxf(rmax[r], sim);
      }
    }
  }
  // max across the 16 lanes that share each row
  for (int r = 0; r < 8; ++r) {
    float v = rmax[r];
    for (int off = 1; off < 16; off <<= 1) v = fmaxf(v, __shfl_xor(v, off, 32));
    rmax[r] = v;
  }
  if (lane16 == 0) {
    for (int r = 0; r < 8; ++r)
      out[((size_t)(rbase + r) * kB + b) * 64 + 16 + dir * 8 + l] = rmax[r];
  }
}

// ---------------------------------------------------------------------------
// Kernel 4: alpha = cos(p,q) per batch — WMMA P·Qᵀ, normalized, to workspace
// ---------------------------------------------------------------------------
__global__ __launch_bounds__(128) void alpha_kernel(
    const float* __restrict__ p, const float* __restrict__ q,
    const float* __restrict__ npl, const float* __restrict__ nql,
    float* __restrict__ alpha, int dir) {
  const int b = blockIdx.x, sblk = blockIdx.y;
  const int dirOff = dir * kDH;
  const int tid = threadIdx.x;
  const int wv = tid >> 5, lane = tid & 31, lane16 = lane & 15, kh = lane >> 4;

  __shared__ float pl[64 * 68];
  __shared__ float ql[128 * 68];

  const int rbase = sblk * 64 + wv * 16 + kh * 8;
  float n1v[8];
  for (int r = 0; r < 8; ++r)
    n1v[r] = npl[(size_t)dir * kS * kB + (rbase + r) * kB + b];

  const int aRowLds = (wv * 16 + lane16) * 68;

  for (int tb = 0; tb < 2; ++tb) {
    v8f acc[8];
    const v8f vzero = {0.f, 0.f, 0.f, 0.f, 0.f, 0.f, 0.f, 0.f};
    for (int tt = 0; tt < 8; ++tt) acc[tt] = vzero;

    for (int kc = 0; kc < 4; ++kc) {
      __syncthreads();
      for (int i = tid; i < 64 * 64; i += 128) {
        const int r = i >> 6, k = i & 63;
        pl[r * 68 + k] =
            p[((size_t)(sblk * 64 + r) * kB + b) * kD2 + dirOff + kc * 64 + k];
      }
      for (int i = tid; i < 128 * 64; i += 128) {
        const int r = i >> 6, k = i & 63;
        ql[r * 68 + k] =
            q[((size_t)(tb * 128 + r) * kB + b) * kD2 + dirOff + kc * 64 + k];
      }
      __syncthreads();
      for (int tt = 0; tt < 8; ++tt) {
        const int bRowLds = (tt * 16 + lane16) * 68;
        v8f c = acc[tt];
        for (int k0 = 0; k0 < 64; k0 += 4) {
          v2f a  = *(const v2f*)&pl[aRowLds + k0 + 2 * kh];
          v2f bv = *(const v2f*)&ql[bRowLds + k0 + 2 * kh];
          c = wmma_f32(a, bv, c);
        }
        acc[tt] = c;
      }
    }
    for (int tt = 0; tt < 8; ++tt) {
      const int t = tb * 128 + tt * 16 + lane16;
      const float n2 = nql[(size_t)dir * kS * kB + t * kB + b];
      for (int r = 0; r < 8; ++r) {
        alpha[(size_t)b * kS * kS + (size_t)(rbase + r) * kS + t] =
            acc[tt][r] / fmaxf(n1v[r] * n2, kEPS);
      }
    }
  }
}

// ---------------------------------------------------------------------------
// Kernel 5: per-row sum and argmax of alpha over t
// ---------------------------------------------------------------------------
__global__ __launch_bounds__(128) void alpha_reduce_kernel(
    const float* __restrict__ alpha, float* __restrict__ rsum,
    int* __restrict__ idxb) {
  const int gid = blockIdx.x * 128 + threadIdx.x;   // 0..32767
  const int b = gid & (kB - 1), s = gid >> 7;
  const float* row = &alpha[(size_t)b * kS * kS + (size_t)s * kS];
  float sum = 0.f, best = -3.4e38f;
  int bi = 0;
  for (int t = 0; t < kS; ++t) {
    const float v = row[t];
    sum += v;
    if (v > best) { best = v; bi = t; }
  }
  rsum[s * kB + b] = sum;
  idxb[s * kB + b] = bi;
}

// ---------------------------------------------------------------------------
// Kernel 6: h_mean = (alpha · Q) / rowsum — WMMA
// ---------------------------------------------------------------------------
__global__ __launch_bounds__(128) void hmean_kernel(
    const float* __restrict__ q, const float* __restrict__ alpha,
    const float* __restrict__ rsum, float* __restrict__ hmean, int dir) {
  const int b = blockIdx.x, sblk = blockIdx.y;
  const int dirOff = dir * kDH;
  const int tid = threadIdx.x;
  const int wv = tid >> 5, lane = tid & 31, lane16 = lane & 15, kh = lane >> 4;

  __shared__ float al[64 * 68];     // [s_local][t_chunk]
  __shared__ float ql2[64 * 132];   // [t_local][d_block]

  const int rbase = sblk * 64 + wv * 16 + kh * 8;
  float rs[8];
  for (int r = 0; r < 8; ++r) rs[r] = rsum[(rbase + r) * kB + b];

  const int aRowLds = (wv * 16 + lane16) * 68;

  for (int dblk = 0; dblk < 2; ++dblk) {
    v8f acc[8];
    const v8f vzero = {0.f, 0.f, 0.f, 0.f, 0.f, 0.f, 0.f, 0.f};
    for (int dt = 0; dt < 8; ++dt) acc[dt] = vzero;

    for (int kc = 0; kc < 4; ++kc) {
      __syncthreads();
      for (int i = tid; i < 64 * 64; i += 128) {   // alpha chunk
        const int r = i >> 6, k = i & 63;
        al[r * 68 + k] =
            alpha[(size_t)b * kS * kS + (size_t)(sblk * 64 + r) * kS + kc * 64 + k];
      }
      for (int i = tid; i < 64 * 128; i += 128) {  // q chunk [t][d]
        const int r = i >> 7, d = i & 127;
        ql2[r * 132 + d] =
            q[((size_t)(kc * 64 + r) * kB + b) * kD2 + dirOff + dblk * 128 + d];
      }
      __syncthreads();
      for (int dt = 0; dt < 8; ++dt) {
        const int n = dt * 16 + lane16;
        v8f c = acc[dt];
        for (int k0 = 0; k0 < 64; k0 += 4) {
          v2f a = *(const v2f*)&al[aRowLds + k0 + 2 * kh];
          v2f bv;
          bv.x = ql2[(k0 + 2 * kh) * 132 + n];
          bv.y = ql2[(k0 + 2 * kh + 1) * 132 + n];
          c = wmma_f32(a, bv, c);
        }
        acc[dt] = c;
      }
    }
    for (int dt = 0; dt < 8; ++dt) {
      const int d = dblk * 128 + dt * 16 + lane16;
      for (int r = 0; r < 8; ++r)
        hmean[((size_t)(rbase + r) * kB + b) * kDH + d] = acc[dt][r] / rs[r];
    }
  }
}

// ---------------------------------------------------------------------------
// Kernel 7: m_att (W4, ch 32+dir*8) and m_maxatt (W5, ch 48+dir*8)
// ---------------------------------------------------------------------------
__global__ __launch_bounds__(256) void att_match_kernel(
    const float* __restrict__ p, const float* __restrict__ q,
    const float* __restrict__ hmean, const int* __restrict__ idxb,
    const float* __restrict__ W, float* __restrict__ out, int dir) {
  const int s = blockIdx.x, b = blockIdx.y;
  const int dirOff = dir * kDH;
  const int tid = threadIdx.x, wv = tid >> 5, lane = tid & 31;
  __shared__ float pr[kDH], hr[kDH], qi[kDH];
  const int idx = idxb[s * kB + b];
  pr[tid] = p[((size_t)s * kB + b) * kD2 + dirOff + tid];
  hr[tid] = hmean[((size_t)s * kB + b) * kDH + tid];
  qi[tid] = q[((size_t)idx * kB + b) * kD2 + dirOff + tid];
  __syncthreads();

  const int l = wv;
  float n4 = 0.f, a4 = 0.f, h4 = 0.f, n5 = 0.f, a5 = 0.f, q5 = 0.f;
  for (int j = 0; j < 8; ++j) {
    const int d = lane + j * 32;
    float w4 = W[(4 * kL + l) * kDH + d]; w4 *= w4;
    float w5 = W[(5 * kL + l) * kDH + d]; w5 *= w5;
    const float pv = pr[d], hv = hr[d], qv = qi[d];
    n4 += pv * hv * w4; a4 += pv * pv * w4; h4 += hv * hv * w4;
    n5 += pv * qv * w5; a5 += pv * pv * w5; q5 += qv * qv * w5;
  }
  n4 = wave_sum(n4); a4 = wave_sum(a4); h4 = wave_sum(h4);
  n5 = wave_sum(n5); a5 = wave_sum(a5); q5 = wave_sum(q5);
  if (lane == 0) {
    float* o = out + ((size_t)s * kB + b) * 64;
    o[32 + dir * 8 + l] = n4 / fmaxf(sqrtf(a4) * sqrtf(h4), kEPS);
    o[48 + dir * 8 + l] = n5 / fmaxf(sqrtf(a5) * sqrtf(q5), kEPS);
  }
}

// ---------------------------------------------------------------------------
extern "C" void kernel_launch(void* const* d_in, const int* in_sizes, int n_in,
                              void* d_out, int out_size, void* d_ws, size_t ws_size,
                              hipStream_t stream) {
  (void)in_sizes; (void)n_in; (void)out_size; (void)ws_size;
  const float* p = (const float*)d_in[0];
  const float* q = (const float*)d_in[1];
  const float* W = (const float*)d_in[2];
  float* out = (float*)d_out;
  float* ws = (float*)d_ws;

  float* alpha = ws + WS_ALPHA;
  float* hmean = ws + WS_HMEAN;
  float* rsum  = ws + WS_RSUM;
  int*   idxb  = (int*)(ws + WS_IDX);
  float* npl   = ws + WS_NPL;
  float* nql   = ws + WS_NQL;
  float* pnw   = ws + WS_PNW;
  float* qnw   = ws + WS_QNW;

  norms_kernel<<<dim3(kS, kB), 256, 0, stream>>>(p, q, W, npl, nql, pnw, qnw);
  full_match_kernel<<<dim3(kS, kB), 256, 0, stream>>>(p, q, W, out);
  maxpool_kernel<<<dim3(kB, kL, 8), 128, 0, stream>>>(p, q, W, pnw, qnw, out);

  for (int dir = 0; dir < 2; ++dir) {
    alpha_kernel<<<dim3(kB, 4), 128, 0, stream>>>(p, q, npl, nql, alpha, dir);
    alpha_reduce_kernel<<<kS, 128, 0, stream>>>(alpha, rsum, idxb);
    hmean_kernel<<<dim3(kB, 4), 128, 0, stream>>>(q, alpha, rsum, hmean, dir);
    att_match_kernel<<<dim3(kS, kB), 256, 0, stream>>>(p, q, hmean, idxb, W, out, dir);
  }
}